// AdaptiveKalmanFilterCXCYWH_63170378990171
// MI455X (gfx1250) — compile-verified
//
#include <hip/hip_runtime.h>
#include <stdint.h>

// AdaptiveKalmanFilterCXCYWH — MI455X (gfx1250)
// Bandwidth-bound (~312MB @ 23.3TB/s => ~13.4us floor). One thread per track,
// fp32 VALU core exploiting F/H/R structure; covariance staged via the CDNA5
// Tensor Data Mover (tensor_load_to_lds) with LDS row padding for bank spread.

typedef unsigned int u32x4 __attribute__((ext_vector_type(4)));
typedef int          i32x8 __attribute__((ext_vector_type(8)));
typedef int          i32x4 __attribute__((ext_vector_type(4)));

#define TPB 256
#define WAVES_PER_BLOCK 8
#define TRACK_LDS_STRIDE 272                    // 64 floats + 16B TDM pad
#define WAVE_LDS_BYTES (32 * TRACK_LDS_STRIDE)  // 8704 B per wave

__global__ __launch_bounds__(TPB) void kf_kernel(
    const float* __restrict__ mean, const float* __restrict__ cov,
    const float* __restrict__ meas, const float* __restrict__ conf,
    float* __restrict__ out_mean, float* __restrict__ out_cov, int B)
{
    __shared__ __align__(16) unsigned char smem[WAVES_PER_BLOCK * WAVE_LDS_BYTES];

    const int tid  = threadIdx.x;
    const int lane = tid & 31;
    const int wv   = __builtin_amdgcn_readfirstlane(tid >> 5);
    const int t0   = blockIdx.x * TPB;
    const int wtrack = t0 + wv * 32;            // first track of this wave

#if __has_builtin(__builtin_amdgcn_tensor_load_to_lds)
    // ---- TDM: DMA 32 tracks x 256B of covariance into LDS, +16B pad per row.
    if (wtrack < B) {
        int nrows = B - wtrack; if (nrows > 32) nrows = 32;
        unsigned long long ga = (unsigned long long)(const void*)(cov + (size_t)wtrack * 64);
        unsigned lbase = (unsigned)(uintptr_t)(&smem[wv * WAVE_LDS_BYTES]); // LDS byte addr (low 32b of generic)
        u32x4 g0;
        g0.x = 1u;                                            // count = 1 (valid user D#)
        g0.y = lbase;                                         // lds_addr [63:32]
        g0.z = (unsigned)ga;                                  // global_addr [95:64]
        g0.w = ((unsigned)(ga >> 32) & 0x01FFFFFFu)           // global_addr [120:96]
             | 0x80000000u;                                   // type = 2 ("image") [127:126]
        i32x8 g1;
        g1.s0 = (int)((2u << 16)                              // data_size = 4B
                    | (1u << 20)                              // pad_enable
                    | (5u << 22)                              // pad_interval: 2^(5+1)=64 DW (256B)
                    | (3u << 25));                            // pad_amount: 3+1 = 4 DW (16B)
        g1.s1 = (int)(64u << 16);                             // tensor_dim0 = 64 elems
        g1.s2 = (int)((unsigned)nrows << 16);                 // tensor_dim1 = nrows
        g1.s3 = (int)(64u << 16);                             // tile_dim0 = 64
        g1.s4 = nrows;                                        // tile_dim1 = nrows (tile_dim2 = 0)
        g1.s5 = 64;                                           // tensor_dim0_stride = 64 elems
        g1.s6 = 0; g1.s7 = 0;
        i32x4 gz  = {0, 0, 0, 0};                             // groups 2/3 unused (2-D tensor)
        i32x8 gz8 = {0, 0, 0, 0, 0, 0, 0, 0};                 // extra group (clang-23 6-arg form)
        __builtin_amdgcn_tensor_load_to_lds(g0, g1, gz, gz, gz8, 0);
    }
  #if __has_builtin(__builtin_amdgcn_s_wait_tensorcnt)
    __builtin_amdgcn_s_wait_tensorcnt(0);
  #else
    __asm__ volatile("s_wait_tensorcnt 0x0" ::: "memory");
  #endif
    __asm__ volatile("" ::: "memory");
#else
    // Fallback staging: same padded layout, regular coalesced loads.
    #pragma unroll
    for (int j = 0; j < 16; ++j) {
        int c = j * 32 + lane;                 // 16B chunk id within wave region
        int trk = c >> 4, off = c & 15;
        if (wtrack + trk < B)
            *(float4*)(&smem[wv * WAVE_LDS_BYTES + trk * TRACK_LDS_STRIDE + off * 16]) =
                *(const float4*)((const char*)(cov + (size_t)wtrack * 64) + (size_t)c * 16);
    }
#endif
    __syncthreads();

    const int t = t0 + tid;
    if (t >= B) return;

    const unsigned char* myb = &smem[wv * WAVE_LDS_BYTES + lane * TRACK_LDS_STRIDE];

    // ---- predict covariance: Pp = F P F^T + diag(q), F = [[I,I],[0,I]] ----
    float Pp[8][8];
    #pragma unroll
    for (int i = 0; i < 4; ++i) {
        float4 a0 = *(const float4*)(myb + i * 32);
        float4 a1 = *(const float4*)(myb + i * 32 + 16);
        float4 b0 = *(const float4*)(myb + (i + 4) * 32);
        float4 b1 = *(const float4*)(myb + (i + 4) * 32 + 16);
        float pi[8] = {a0.x, a0.y, a0.z, a0.w, a1.x, a1.y, a1.z, a1.w};
        float pv[8] = {b0.x, b0.y, b0.z, b0.w, b1.x, b1.y, b1.z, b1.w};
        #pragma unroll
        for (int j = 0; j < 4; ++j) {
            float vv = pv[j + 4];
            Pp[i][j]         = pi[j] + pi[j + 4] + pv[j] + vv;
            Pp[i][j + 4]     = pi[j + 4] + vv;
            Pp[i + 4][j]     = pv[j] + vv;
            Pp[i + 4][j + 4] = vv;
        }
    }
    #pragma unroll
    for (int i = 0; i < 4; ++i) { Pp[i][i] += 0.01f; Pp[i + 4][i + 4] += 1e-4f; }

    // ---- predict mean ----
    const float4* m4 = (const float4*)(mean + (size_t)t * 8);
    float4 m0 = m4[0], m1 = m4[1];
    float mp[8] = { m0.x + m1.x, m0.y + m1.y, m0.z + m1.z, m0.w + m1.w,
                    m1.x, m1.y, m1.z, m1.w };

    // ---- measurement noise R = rr*I (both std constants are 1/20) ----
    float4 z = ((const float4*)meas)[t];
    float h   = fmaxf(z.w, 1.0f);
    float cw  = 1.0f / fmaxf(conf[t], 1e-6f);
    float std = fmaxf(h * 0.05f * cw, 0.1f);
    float rr  = std * std + 1e-6f;              // R diagonal
    float y0i = z.x - mp[0], y1i = z.y - mp[1], y2i = z.z - mp[2], y3i = z.w - mp[3];

    // ---- S = Pp[0:4,0:4] + (rr + 1e-6) I ; Cholesky 4x4 (lower) ----
    float sadd = rr + 1e-6f;
    float S00 = Pp[0][0] + sadd, S10 = Pp[1][0], S11 = Pp[1][1] + sadd;
    float S20 = Pp[2][0], S21 = Pp[2][1], S22 = Pp[2][2] + sadd;
    float S30 = Pp[3][0], S31 = Pp[3][1], S32 = Pp[3][2], S33 = Pp[3][3] + sadd;

    float L00 = sqrtf(S00), i0 = 1.0f / L00;
    float L10 = S10 * i0, L20 = S20 * i0, L30 = S30 * i0;
    float t11 = S11 - L10 * L10;                 float L11 = sqrtf(t11), i1 = 1.0f / L11;
    float L21 = (S21 - L20 * L10) * i1;
    float L31 = (S31 - L30 * L10) * i1;
    float t22 = S22 - L20 * L20 - L21 * L21;     float L22 = sqrtf(t22), i2 = 1.0f / L22;
    float L32 = (S32 - L30 * L20 - L31 * L21) * i2;
    float t33 = S33 - L30 * L30 - L31 * L31 - L32 * L32;
    float i3 = 1.0f / sqrtf(t33);

    // ---- K = Pp[:,0:4] * S^-1 via two triangular solves per row ----
    float K[8][4];
    #pragma unroll
    for (int g = 0; g < 8; ++g) {
        float c0 = Pp[g][0], c1 = Pp[g][1], c2 = Pp[g][2], c3 = Pp[g][3];
        float y0 = c0 * i0;
        float y1 = (c1 - L10 * y0) * i1;
        float y2 = (c2 - L20 * y0 - L21 * y1) * i2;
        float y3 = (c3 - L30 * y0 - L31 * y1 - L32 * y2) * i3;
        float x3 = y3 * i3;
        float x2 = (y2 - L32 * x3) * i2;
        float x1 = (y1 - L21 * x2 - L31 * x3) * i1;
        float x0 = (y0 - L10 * x1 - L20 * x2 - L30 * x3) * i0;
        K[g][0] = x0; K[g][1] = x1; K[g][2] = x2; K[g][3] = x3;
    }

    // ---- mean_new = mp + K * innovation ----
    float mn[8];
    #pragma unroll
    for (int i = 0; i < 8; ++i)
        mn[i] = mp[i] + K[i][0] * y0i + K[i][1] * y1i + K[i][2] * y2i + K[i][3] * y3i;

    float* om = out_mean + (size_t)t * 8;
    ((float4*)om)[0] = make_float4(mn[0], mn[1], mn[2], mn[3]);
    ((float4*)om)[1] = make_float4(mn[4], mn[5], mn[6], mn[7]);

    // ---- T = (I - K H) Pp, in place (M = I with first 4 cols minus K) ----
    #pragma unroll
    for (int i = 4; i < 8; ++i)                  // rows 4..7 first (read rows 0..3)
        #pragma unroll
        for (int j = 0; j < 8; ++j)
            Pp[i][j] -= K[i][0] * Pp[0][j] + K[i][1] * Pp[1][j]
                      + K[i][2] * Pp[2][j] + K[i][3] * Pp[3][j];
    #pragma unroll
    for (int j = 0; j < 8; ++j) {                // rows 0..3 column-wise in place
        float a = Pp[0][j], b = Pp[1][j], c = Pp[2][j], d = Pp[3][j];
        Pp[0][j] = a - (K[0][0] * a + K[0][1] * b + K[0][2] * c + K[0][3] * d);
        Pp[1][j] = b - (K[1][0] * a + K[1][1] * b + K[1][2] * c + K[1][3] * d);
        Pp[2][j] = c - (K[2][0] * a + K[2][1] * b + K[2][2] * c + K[2][3] * d);
        Pp[3][j] = d - (K[3][0] * a + K[3][1] * b + K[3][2] * c + K[3][3] * d);
    }

    // ---- U = T M^T + rr * K K^T, in place per row ----
    #pragma unroll
    for (int i = 0; i < 8; ++i) {
        float u0 = rr * K[i][0] - Pp[i][0];
        float u1 = rr * K[i][1] - Pp[i][1];
        float u2 = rr * K[i][2] - Pp[i][2];
        float u3 = rr * K[i][3] - Pp[i][3];
        #pragma unroll
        for (int j = 0; j < 8; ++j)
            Pp[i][j] += u0 * K[j][0] + u1 * K[j][1] + u2 * K[j][2] + u3 * K[j][3];
    }

    // ---- symmetrize + store ----
    float* oc = out_cov + (size_t)t * 64;
    #pragma unroll
    for (int i = 0; i < 8; ++i) {
        float4 r0, r1;
        r0.x = 0.5f * (Pp[i][0] + Pp[0][i]);
        r0.y = 0.5f * (Pp[i][1] + Pp[1][i]);
        r0.z = 0.5f * (Pp[i][2] + Pp[2][i]);
        r0.w = 0.5f * (Pp[i][3] + Pp[3][i]);
        r1.x = 0.5f * (Pp[i][4] + Pp[4][i]);
        r1.y = 0.5f * (Pp[i][5] + Pp[5][i]);
        r1.z = 0.5f * (Pp[i][6] + Pp[6][i]);
        r1.w = 0.5f * (Pp[i][7] + Pp[7][i]);
        ((float4*)oc)[i * 2]     = r0;
        ((float4*)oc)[i * 2 + 1] = r1;
    }
}

extern "C" void kernel_launch(void* const* d_in, const int* in_sizes, int n_in,
                              void* d_out, int out_size, void* d_ws, size_t ws_size,
                              hipStream_t stream) {
    const float* mean = (const float*)d_in[0];
    const float* cov  = (const float*)d_in[1];
    const float* meas = (const float*)d_in[2];
    const float* conf = (const float*)d_in[3];
    const int B = in_sizes[3];                  // confidence: one element per track
    float* out_mean = (float*)d_out;
    float* out_cov  = out_mean + (size_t)B * 8; // tuple order: mean_new, cov_new
    const int grid = (B + TPB - 1) / TPB;
    kf_kernel<<<grid, TPB, 0, stream>>>(mean, cov, meas, conf, out_mean, out_cov, B);
    (void)d_ws; (void)ws_size; (void)n_in; (void)out_size;
}